// EvolutionModel_69827578298857
// MI455X (gfx1250) — compile-verified
//
#include <hip/hip_runtime.h>
#include <math.h>

// ---------------------------------------------------------------------------
// EvolutionModel: tetra-mesh gradient-index ray tracer for MI455X (gfx1250).
// Point location = 4Tx4 * 4xP f32 GEMM on V_WMMA_F32_16X16X4_F32, with the
// A matrix pre-swizzled into per-lane fragment order (b128 loads, 4x unroll).
// ---------------------------------------------------------------------------

typedef float v2f __attribute__((ext_vector_type(2)));
typedef float v4f __attribute__((ext_vector_type(4)));
typedef float v8f __attribute__((ext_vector_type(8)));

#define MAX_STEPS 64
#define TWO_PI 6.283185307179586f
#define INT_SENTINEL 0x7FFFFFFF

struct F3 { float x, y, z; };
__device__ __forceinline__ F3 mk3(float a, float b, float c) { F3 r; r.x=a; r.y=b; r.z=c; return r; }
__device__ __forceinline__ F3 add3(F3 a, F3 b) { return mk3(a.x+b.x, a.y+b.y, a.z+b.z); }
__device__ __forceinline__ F3 sub3(F3 a, F3 b) { return mk3(a.x-b.x, a.y-b.y, a.z-b.z); }
__device__ __forceinline__ F3 scl3(F3 a, float s) { return mk3(a.x*s, a.y*s, a.z*s); }
__device__ __forceinline__ float dot3(F3 a, F3 b) { return a.x*b.x + a.y*b.y + a.z*b.z; }
__device__ __forceinline__ F3 cross3(F3 a, F3 b) {
    return mk3(a.y*b.z - a.z*b.y, a.z*b.x - a.x*b.z, a.x*b.y - a.y*b.x);
}
__device__ __forceinline__ F3 ld3(const float* p, int i) { return mk3(p[3*i], p[3*i+1], p[3*i+2]); }
__device__ __forceinline__ void st3(float* p, F3 v) { p[0]=v.x; p[1]=v.y; p[2]=v.z; }
__device__ __forceinline__ float wrap2pi(float x) {
    float r = fmodf(x, TWO_PI);
    if (r < 0.f) r += TWO_PI;
    return r;
}

// ---------------------------------------------------------------------------
// Kernel A: per-tet precompute.
//  Logical A matrix rows (4 per tet): [Einv_i | -(Einv v0)_i], row3 = col sums
//  so D = A * [p;1] = [lam1, lam2, lam3, lam1+lam2+lam3].
//  Stored PRE-SWIZZLED in WMMA 16x4-f32 fragment order, two tiles per 128-f
//  block: addr = pair*128 + lane*4 + (tile&1)*2 + k%2, lane = (k/2)*16 + m.
//  abn[t*8]: a, b.xyz, n.xyz, |b|   (affine fit n(x) = a + b.x)
// ---------------------------------------------------------------------------
__global__ void precompute_kernel(const float* __restrict__ pos,
                                  const float* __restrict__ n_index,
                                  const int*   __restrict__ tetra,
                                  float* __restrict__ tetAsw,
                                  float* __restrict__ abn,
                                  int T) {
    int t = blockIdx.x * blockDim.x + threadIdx.x;
    if (t >= T) return;
    int i0 = tetra[4*t+0], i1 = tetra[4*t+1], i2 = tetra[4*t+2], i3 = tetra[4*t+3];
    F3 v0 = ld3(pos, i0), v1 = ld3(pos, i1), v2 = ld3(pos, i2), v3 = ld3(pos, i3);
    F3 e1 = sub3(v1, v0), e2 = sub3(v2, v0), e3 = sub3(v3, v0);
    F3 c23 = cross3(e2, e3), c31 = cross3(e3, e1), c12 = cross3(e1, e2);
    float det = dot3(e1, c23);
    float inv = 1.0f / det;
    F3 r0 = scl3(c23, inv), r1 = scl3(c31, inv), r2 = scl3(c12, inv); // rows of Einv
    float o0 = -dot3(r0, v0), o1 = -dot3(r1, v0), o2 = -dot3(r2, v0);

    // rows[i][k] for this tet
    float rows[4][4] = {
        { r0.x, r0.y, r0.z, o0 },
        { r1.x, r1.y, r1.z, o1 },
        { r2.x, r2.y, r2.z, o2 },
        { r0.x+r1.x+r2.x, r0.y+r1.y+r2.y, r0.z+r1.z+r2.z, o0+o1+o2 }
    };
    const int tile = t >> 2;          // 4 tets per 16-row tile
    const int sub  = tile & 1;        // position within tile-pair
    const size_t base = (size_t)(tile >> 1) * 128 + (size_t)sub * 2;
    #pragma unroll
    for (int i = 0; i < 4; ++i) {
        int m = (t & 3) * 4 + i;      // row within tile
        #pragma unroll
        for (int k = 0; k < 4; ++k) {
            int lane = (k >> 1) * 16 + m;
            tetAsw[base + (size_t)lane * 4 + (k & 1)] = rows[i][k];
        }
    }

    float n0 = n_index[i0], n1 = n_index[i1], n2 = n_index[i2], n3 = n_index[i3];
    F3 dn = mk3(n1-n0, n2-n0, n3-n0);
    // b = Einv^T dn  (solves E^T b = dn: affine interpolation of n at verts)
    F3 b = mk3(r0.x*dn.x + r1.x*dn.y + r2.x*dn.z,
               r0.y*dn.x + r1.y*dn.y + r2.y*dn.z,
               r0.z*dn.x + r1.z*dn.y + r2.z*dn.z);
    float a  = n0 - dot3(b, v0);
    float nb = sqrtf(dot3(b, b));
    F3 nn = scl3(b, 1.0f / nb);
    float* o = abn + t * 8;
    o[0]=a; o[1]=b.x; o[2]=b.y; o[3]=b.z; o[4]=nn.x; o[5]=nn.y; o[6]=nn.z; o[7]=nb;
}

__global__ void init_t0_kernel(int* __restrict__ t0buf, int n) {
    int i = blockIdx.x * blockDim.x + threadIdx.x;
    if (i < n) t0buf[i] = INT_SENTINEL;
}

// ---------------------------------------------------------------------------
// Kernel B: brute-force point location as f32 WMMA GEMM.
// Per quad (4 tiles = 16 tets): 2x global_load_b128 + 4x wmma + checks.
// ---------------------------------------------------------------------------
__device__ __forceinline__ void check_tets(const v8f acc, int tet0, int T, int& best) {
    {
        float l1 = acc[0], l2 = acc[1], l3 = acc[2], s = acc[3];
        bool ok = (l1 >= 0.f) & (l2 >= 0.f) & (l3 >= 0.f) &
                  (l1 <= 1.f) & (l2 <= 1.f) & (l3 <= 1.f) & (s <= 1.f);
        if (ok && tet0 < T && tet0 < best) best = tet0;
    }
    {
        float l1 = acc[4], l2 = acc[5], l3 = acc[6], s = acc[7];
        int tt = tet0 + 1;
        bool ok = (l1 >= 0.f) & (l2 >= 0.f) & (l3 >= 0.f) &
                  (l1 <= 1.f) & (l2 <= 1.f) & (l3 <= 1.f) & (s <= 1.f);
        if (ok && tt < T && tt < best) best = tt;
    }
}

__global__ void locate_kernel(const float* __restrict__ r0,
                              const float* __restrict__ tetAsw,
                              int* __restrict__ t0buf,
                              int nrays, int T, int numQuads) {
    const int lane  = threadIdx.x & 31;
    const int half  = lane >> 4;     // 0: K=0,1 / D rows 0..7 ; 1: K=2,3 / rows 8..15
    const int n     = lane & 15;     // ray column within tile
    const int ray   = blockIdx.x * 16 + n;

    // B fragment: column [px, py, pz, 1]
    float px = 0.f, py = 0.f, pz = 0.f;
    if (ray < nrays) { px = r0[3*ray]; py = r0[3*ray+1]; pz = r0[3*ray+2]; }
    v2f bfrag;
    if (half == 0) { bfrag.x = px; bfrag.y = py; }
    else           { bfrag.x = pz; bfrag.y = 1.0f; }

    const int wavesPerBlk = blockDim.x >> 5;
    const int wave   = blockIdx.y * wavesPerBlk + (threadIdx.x >> 5);
    const int stride = gridDim.y * wavesPerBlk;

    int best = INT_SENTINEL;
    for (int q = wave; q < numQuads; q += stride) {
        // prefetch next quad's fragment block (1KB/wave) into cache
        int qn = q + stride;
        if (qn < numQuads)
            __builtin_prefetch(tetAsw + (size_t)(2 * qn) * 128 + lane * 4, 0, 0);

        // two b128 loads cover 4 tiles (16 tets) of A fragments for this lane
        const v4f a01 = *(const v4f*)(tetAsw + (size_t)(2*q  ) * 128 + lane * 4);
        const v4f a23 = *(const v4f*)(tetAsw + (size_t)(2*q+1) * 128 + lane * 4);

        v2f f0; f0.x = a01.x; f0.y = a01.y;
        v2f f1; f1.x = a01.z; f1.y = a01.w;
        v2f f2; f2.x = a23.x; f2.y = a23.y;
        v2f f3; f3.x = a23.z; f3.y = a23.w;

        v8f z = {0.f,0.f,0.f,0.f,0.f,0.f,0.f,0.f};
        v8f d0 = __builtin_amdgcn_wmma_f32_16x16x4_f32(false, f0, false, bfrag, (short)0, z, false, false);
        v8f d1 = __builtin_amdgcn_wmma_f32_16x16x4_f32(false, f1, false, bfrag, (short)0, z, false, false);
        v8f d2 = __builtin_amdgcn_wmma_f32_16x16x4_f32(false, f2, false, bfrag, (short)0, z, false, false);
        v8f d3 = __builtin_amdgcn_wmma_f32_16x16x4_f32(false, f3, false, bfrag, (short)0, z, false, false);

        const int tbase = q * 16;                      // 4 tets per tile
        check_tets(d0, tbase +  0 + half * 2, T, best);
        check_tets(d1, tbase +  4 + half * 2, T, best);
        check_tets(d2, tbase +  8 + half * 2, T, best);
        check_tets(d3, tbase + 12 + half * 2, T, best);
    }
    // combine the two halves covering the same ray column
    int other = __shfl_xor(best, 16, 32);
    best = min(best, other);
    if (lane < 16 && ray < nrays && best != INT_SENTINEL)
        atomicMin(&t0buf[ray], best);
}

// ---------------------------------------------------------------------------
// Kernel C: circular-arc ray evolution, one thread per ray.
// ---------------------------------------------------------------------------
__global__ void evolve_kernel(const float* __restrict__ r0,
                              const float* __restrict__ m0,
                              const float* __restrict__ pos,
                              const int*   __restrict__ tetra_face,
                              const int*   __restrict__ face_vertex,
                              const int*   __restrict__ face_tetra,
                              const float* __restrict__ abn,
                              const int*   __restrict__ t0buf,
                              const int*   __restrict__ n_steps_p,
                              float* __restrict__ rhist,
                              float* __restrict__ cum,
                              int nrays) {
    int ray = blockIdx.x * blockDim.x + threadIdx.x;
    if (ray >= nrays) return;
    int nst = *n_steps_p; if (nst > MAX_STEPS) nst = MAX_STEPS; if (nst < 0) nst = 0;

    F3 r = ld3(r0, ray);
    F3 m = ld3(m0, ray);
    int tb = t0buf[ray];
    int t = (tb == INT_SENTINEL) ? -1 : tb;

    const int K = MAX_STEPS + 1;
    st3(rhist + (ray * K + 0) * 3, r);
    cum[ray * K + 0] = 0.f;
    float cd = 0.f;

    for (int s = 0; s < nst; ++s) {
        bool dead = (t < 0);
        int ti = dead ? 0 : t;
        const float* ab = abn + ti * 8;
        float a = ab[0];
        F3 n = mk3(ab[4], ab[5], ab[6]);
        float normb = ab[7];

        F3 mn = cross3(m, n);
        F3 q  = scl3(mn, 1.0f / sqrtf(dot3(mn, mn)));
        F3 nq = cross3(n, q);
        float mnd  = dot3(m, n);
        float mdnq = dot3(m, nq);
        float coef = dot3(r, n) + a / normb;
        F3 dir = sub3(n, scl3(nq, mnd / mdnq));
        F3 rc  = sub3(r, scl3(dir, coef));
        F3 R   = sub3(rc, r);
        float Rn = sqrtf(dot3(R, R));

        float bestphi = 1e30f;
        int   bestf   = 0;
        #pragma unroll
        for (int f = 0; f < 4; ++f) {
            int fid = tetra_face[4*ti + f];
            int a0 = face_vertex[3*fid+0], a1 = face_vertex[3*fid+1], a2 = face_vertex[3*fid+2];
            F3 vi = ld3(pos, a0), vj = ld3(pos, a1), vk = ld3(pos, a2);
            F3 ML = cross3(sub3(vj, vi), sub3(vk, vi));
            float QL = -dot3(vi, ML);
            float c1 = -dot3(ML, R);
            float c2 = Rn * dot3(ML, m);
            float c3 = dot3(ML, rc) + QL;
            float disc = c1*c1 + c2*c2 - c3*c3;
            bool valid = (disc >= 0.f);
            float sq = sqrtf(valid ? disc : 1.f);
            float denom = c1 - c3;
            if (denom == 0.f) denom = 1e-20f;
            float phi1 = wrap2pi(2.f * atanf((c2 + sq) / denom));
            float phi2 = wrap2pi(2.f * atanf((c2 - sq) / denom));
            float phiR = valid ? fminf(phi1, phi2) : 10.f;
            if (!(phiR == phiR)) phiR = 10.f;      // nan_to_num(nan=10)
            if (phiR < bestphi) { bestphi = phiR; bestf = f; }
        }

        float phiE = bestphi * 1.01f;
        float cE = cosf(phiE), sE = sinf(phiE);
        F3 re = add3(sub3(rc, scl3(R, cE)), scl3(m, Rn * sE));
        F3 me = add3(scl3(m, cE), scl3(R, sE / Rn));

        int hf = tetra_face[4*ti + bestf];
        int f0 = face_tetra[2*hf + 0], f1 = face_tetra[2*hf + 1];
        int nt = (f0 == ti) ? f1 : f0;

        F3 dr = sub3(r, re);
        float dist = sqrtf(dot3(dr, dr));

        if (dead) { nt = -1; re = r; me = m; dist = 0.f; }

        cd += dist;
        st3(rhist + (ray * K + s + 1) * 3, re);
        cum[ray * K + s + 1] = cd;
        r = re; m = me; t = nt;
    }
}

// ---------------------------------------------------------------------------
// Kernel D: sample rays along arclength (searchsorted + lerp).
// ---------------------------------------------------------------------------
__global__ void sample_kernel(const float* __restrict__ rhist,
                              const float* __restrict__ cum,
                              const int*   __restrict__ n_steps_p,
                              float* __restrict__ out,
                              int nrays, int nsamp) {
    int tid = blockIdx.x * blockDim.x + threadIdx.x;
    if (tid >= nrays * nsamp) return;
    int ray = tid / nsamp;
    int s   = tid - ray * nsamp;
    int nst = *n_steps_p; if (nst > MAX_STEPS) nst = MAX_STEPS; if (nst < 0) nst = 0;
    int K = nst + 1;
    const int STRIDE = MAX_STEPS + 1;

    float z = (nsamp > 1) ? (0.1f + 0.9f * (float)s / (float)(nsamp - 1)) : 0.1f;
    const float* d = cum + ray * STRIDE;

    int idx = 0;                                  // searchsorted 'left'
    while (idx < K && d[idx] < z) ++idx;
    if (idx < 1) idx = 1;
    if (idx > K - 1) idx = K - 1;

    float dlo = d[idx - 1], dhi = d[idx];
    float w = (z - dlo) / (dhi - dlo + 1e-8f);
    const float* rl = rhist + (ray * STRIDE + idx - 1) * 3;
    const float* rh = rhist + (ray * STRIDE + idx) * 3;
    out[3*tid + 0] = rl[0] + w * (rh[0] - rl[0]);
    out[3*tid + 1] = rl[1] + w * (rh[1] - rl[1]);
    out[3*tid + 2] = rl[2] + w * (rh[2] - rl[2]);
}

// ---------------------------------------------------------------------------
extern "C" void kernel_launch(void* const* d_in, const int* in_sizes, int n_in,
                              void* d_out, int out_size, void* d_ws, size_t ws_size,
                              hipStream_t stream) {
    const float* r0          = (const float*)d_in[0];
    const float* m0          = (const float*)d_in[1];
    const float* pos         = (const float*)d_in[2];
    const float* n_index     = (const float*)d_in[3];
    const int*   tetra       = (const int*)  d_in[4];
    const int*   tetra_face  = (const int*)  d_in[5];
    const int*   face_vertex = (const int*)  d_in[6];
    const int*   face_tetra  = (const int*)  d_in[7];
    const int*   n_steps_p   = (const int*)  d_in[8];
    float* out = (float*)d_out;

    const int nrays = in_sizes[0] / 3;
    const int T     = in_sizes[4] / 4;
    const int nsamp = out_size / (nrays * 3);

    const int numTiles    = (T + 3) >> 2;           // 16-row WMMA tiles (4 tets each)
    const int paddedTiles = (numTiles + 3) & ~3;    // quad granularity
    const int numQuads    = paddedTiles >> 2;

    // Workspace layout (floats/ints, 4B units)
    float* ws = (float*)d_ws;
    size_t off = 0;
    float* tetAsw = ws + off;  off += (size_t)paddedTiles * 64;   // swizzled A frags
    float* abn    = ws + off;  off += (size_t)T * 8;
    int*   t0b    = (int*)(ws + off); off += (size_t)nrays;
    float* rh     = ws + off;  off += (size_t)nrays * (MAX_STEPS + 1) * 3;
    float* cm     = ws + off;  off += (size_t)nrays * (MAX_STEPS + 1);
    (void)ws_size; (void)n_in;

    // A: per-tet precompute + t0 init (same stream => ordered before B)
    init_t0_kernel<<<(nrays + 255) / 256, 256, 0, stream>>>(t0b, nrays);
    precompute_kernel<<<(T + 255) / 256, 256, 0, stream>>>(pos, n_index, tetra, tetAsw, abn, T);

    // B: WMMA point location. grid.x = ray tiles, grid.y = tet-range chunks.
    {
        dim3 grid((nrays + 15) / 16, 16);
        locate_kernel<<<grid, 256, 0, stream>>>(r0, tetAsw, t0b, nrays, T, numQuads);
    }

    // C: evolution (one thread per ray)
    evolve_kernel<<<(nrays + 255) / 256, 256, 0, stream>>>(
        r0, m0, pos, tetra_face, face_vertex, face_tetra,
        abn, t0b, n_steps_p, rh, cm, nrays);

    // D: sampling
    {
        int nthreads = nrays * nsamp;
        sample_kernel<<<(nthreads + 255) / 256, 256, 0, stream>>>(
            rh, cm, n_steps_p, out, nrays, nsamp);
    }
}